// MaskHeatmapLoss_1657857376806
// MI455X (gfx1250) — compile-verified
//
#include <hip/hip_runtime.h>
#include <hip/hip_bf16.h>

typedef __attribute__((ext_vector_type(2))) float v2f;
typedef __attribute__((ext_vector_type(8))) float v8f;

#define B_ 16
#define P_ 20
#define K_ 17
#define H_ 160
#define W_ 160
#define HW_ (H_ * W_)              // 25600
#define CHUNKS_ 5
#define CHUNK_Q_ (HW_ / CHUNKS_ / 4)  // 1280 float4 per chunk
#define POS_THR_ 0.01f

// ---------------------------------------------------------------------------
// Kernel 1: per-(b,p) expanded bounding boxes + active-joint bitmask.
// activeMask bit k set  <=>  person p (with >=1 visible joint) constrains
// heatmap channel k (i.e. joint k invisible for this person).
// ---------------------------------------------------------------------------
__global__ void boxes_kernel(const float* __restrict__ joints,
                             float4* __restrict__ boxes,
                             unsigned* __restrict__ activeMask) {
  int idx = blockIdx.x * blockDim.x + threadIdx.x;
  if (idx >= B_ * P_) return;
  const float* j = joints + (size_t)idx * K_ * 3;

  float minx = INFINITY, miny = INFINITY;
  float maxx = -INFINITY, maxy = -INFINITY;
  unsigned visMask = 0u;
  for (int k = 0; k < K_; ++k) {
    float x = j[3 * k + 0];
    float y = j[3 * k + 1];
    float v = j[3 * k + 2];
    if (v > 0.0f) {
      visMask |= (1u << k);
      minx = fminf(minx, x);  maxx = fmaxf(maxx, x);
      miny = fminf(miny, y);  maxy = fmaxf(maxy, y);
    }
  }
  float wd = fmaxf(maxx - minx, 1.0f);
  float hd = fmaxf(maxy - miny, 1.0f);
  float cx = 0.5f * (maxx + minx);
  float cy = 0.5f * (maxy + miny);
  // bb_WH = max(bb_WH, bb_WH[::-1] / HW_RATIO), HW_RATIO = 2
  float w2 = fmaxf(wd, hd * 0.5f);
  float h2 = fmaxf(hd, wd * 0.5f);
  float hx = 0.8f * w2;   // (0.5 + EXPANSION) = 0.8
  float hy = 0.8f * h2;
  float4 bx;
  bx.x = rintf(cx - hx);  // TLx   (rintf = round-half-even, matches jnp.round)
  bx.y = rintf(cy - hy);  // TLy
  bx.z = rintf(cx + hx);  // BRx
  bx.w = rintf(cy + hy);  // BRy
  boxes[idx] = bx;
  activeMask[idx] = (visMask != 0u) ? ((~visMask) & ((1u << K_) - 1u)) : 0u;
}

// ---------------------------------------------------------------------------
// Kernel 2: streaming masked-MSE over one (b,k) plane chunk.
// 256 threads, 5 float4 each (5120 px). Wave sums via V_WMMA_F32_16X16X4_F32.
// ---------------------------------------------------------------------------
__global__ __launch_bounds__(256) void loss_kernel(
    const float* __restrict__ pred, const float* __restrict__ gt,
    const float* __restrict__ masks,
    const float4* __restrict__ boxes, const unsigned* __restrict__ activeMask,
    float* __restrict__ partials) {
  __shared__ float4 sbox[P_];
  __shared__ int snum;
  __shared__ float waveSum[8];

  const int bk = blockIdx.x;          // b*K + k
  const int b = bk / K_;
  const int k = bk % K_;
  const int chunk = blockIdx.y;

  if (threadIdx.x == 0) {
    int n = 0;
    for (int p = 0; p < P_; ++p) {
      unsigned am = activeMask[b * P_ + p];
      if ((am >> k) & 1u) sbox[n++] = boxes[b * P_ + p];
    }
    snum = n;
  }
  __syncthreads();
  const int nbox = snum;

  const size_t plane = (size_t)bk * HW_;
  const float4* p4 = (const float4*)(pred + plane);
  const float4* g4 = (const float4*)(gt + plane);
  const float4* m4 = (const float4*)(masks + (size_t)b * HW_);
  const int qbase = chunk * CHUNK_Q_;

  float acc = 0.0f;
#pragma unroll
  for (int jj = 0; jj < 5; ++jj) {
    const int q = qbase + (int)threadIdx.x + jj * 256;
    if (jj < 4) {
      __builtin_prefetch(&p4[q + 256], 0, 0);
      __builtin_prefetch(&g4[q + 256], 0, 0);
    }
    const float4 pv = p4[q];
    const float4 gv = g4[q];
    const float4 mv = m4[q];

    const int idx = q * 4;                 // 160 % 4 == 0 -> same row
    const float fy = (float)(idx / W_);
    const float fx = (float)(idx % W_);

    bool in0 = false, in1 = false, in2 = false, in3 = false;
    for (int i = 0; i < nbox; ++i) {
      const float4 bxv = sbox[i];
      if (fy >= bxv.y && fy <= bxv.w) {     // inside = !outside
        in0 |= (fx        >= bxv.x) && (fx        <= bxv.z);
        in1 |= (fx + 1.0f >= bxv.x) && (fx + 1.0f <= bxv.z);
        in2 |= (fx + 2.0f >= bxv.x) && (fx + 2.0f <= bxv.z);
        in3 |= (fx + 3.0f >= bxv.x) && (fx + 3.0f <= bxv.z);
      }
    }
    // mask = pred>=thr ? 1 : (insideAnyBox ? 0 : masks)
    float d, m;
    d = pv.x - gv.x; m = (pv.x >= POS_THR_) ? 1.0f : (in0 ? 0.0f : mv.x); acc += d * d * m;
    d = pv.y - gv.y; m = (pv.y >= POS_THR_) ? 1.0f : (in1 ? 0.0f : mv.y); acc += d * d * m;
    d = pv.z - gv.z; m = (pv.z >= POS_THR_) ? 1.0f : (in2 ? 0.0f : mv.z); acc += d * d * m;
    d = pv.w - gv.w; m = (pv.w >= POS_THR_) ? 1.0f : (in3 ? 0.0f : mv.w); acc += d * d * m;
  }

  // ---- wave32 reduction via V_WMMA_F32_16X16X4_F32 ----
  // A (16x4, 2 VGPRs): VGPR0 = acc (lanes 0-15 -> A[m][0], lanes 16-31 -> A[m][2]),
  // VGPR1 = 0. B = all ones => D[m][n] = rowsum(A,m) = acc[m] + acc[m+16].
  // Sum of 8 D VGPRs per lane: lanes 0-15 get rows 0-7, lanes 16-31 rows 8-15.
  v2f a;  a.x = acc;  a.y = 0.0f;
  v2f bo; bo.x = 1.0f; bo.y = 1.0f;
  v8f c = {};
  v8f d8 = __builtin_amdgcn_wmma_f32_16x16x4_f32(
      /*neg_a=*/false, a, /*neg_b=*/false, bo,
      /*c_mod=*/(short)0, c, /*reuse_a=*/false, /*reuse_b=*/false);
  float s = d8[0] + d8[1] + d8[2] + d8[3] + d8[4] + d8[5] + d8[6] + d8[7];
  float wtot = s + __shfl(s, 16, 32);      // lane0: rows0-7 + rows8-15 = wave sum

  const int lane = threadIdx.x & 31;
  const int wave = threadIdx.x >> 5;
  if (lane == 0) waveSum[wave] = wtot;
  __syncthreads();
  if (threadIdx.x == 0) {
    float r = 0.0f;
    for (int w = 0; w < 8; ++w) r += waveSum[w];   // fixed order: deterministic
    partials[bk * CHUNKS_ + chunk] = r;
  }
}

// ---------------------------------------------------------------------------
// Kernel 3: deterministic final reduce: 85 partials per batch, fixed order.
// ---------------------------------------------------------------------------
__global__ void finalize_kernel(const float* __restrict__ partials,
                                float* __restrict__ out) {
  int b = threadIdx.x;
  if (b < B_) {
    float s = 0.0f;
    const float* p = partials + (size_t)b * K_ * CHUNKS_;
    for (int i = 0; i < K_ * CHUNKS_; ++i) s += p[i];
    out[b] = s * (1.0f / (float)(K_ * H_ * W_));
  }
}

extern "C" void kernel_launch(void* const* d_in, const int* in_sizes, int n_in,
                              void* d_out, int out_size, void* d_ws, size_t ws_size,
                              hipStream_t stream) {
  const float* hm     = (const float*)d_in[0];  // (B,K,H,W)
  const float* joints = (const float*)d_in[1];  // (B,P,K,3)
  const float* masks  = (const float*)d_in[2];  // (B,H,W)
  const float* gt     = (const float*)d_in[3];  // (B,K,H,W)
  float* out = (float*)d_out;

  char* ws = (char*)d_ws;
  float4*   boxes = (float4*)ws;                  // 320 * 16 B = 5120
  unsigned* amask = (unsigned*)(ws + 5120);       // 320 *  4 B = 1280
  float*    parts = (float*)(ws + 6400);          // 272*5 * 4 B = 5440

  boxes_kernel<<<1, B_ * P_, 0, stream>>>(joints, boxes, amask);
  loss_kernel<<<dim3(B_ * K_, CHUNKS_), 256, 0, stream>>>(
      hm, gt, masks, boxes, amask, parts);
  finalize_kernel<<<1, 32, 0, stream>>>(parts, out);
}